// SparseRecursiveLinear_11175504904589
// MI455X (gfx1250) — compile-verified
//
#include <hip/hip_runtime.h>
#include <stdint.h>

// ---------------- problem constants (from reference) ----------------
#define B_ROWS   8192
#define F0       2048
#define E_ROWS   512
#define OUT_DIM  2048
#define NNZ_E    8192
#define NNZ_M    16384
#define H_WIDTH  (F0 + 3*E_ROWS)   // 3584
#define BTILE    16

// ---------------- workspace layout (bytes) ----------------
#define RP_STRIDE_I   2560                              // ints per matrix (row_ptr)
#define OFF_RP        0
#define OFF_CUR       (4*RP_STRIDE_I*4)                 // 40960
#define OFF_KIDX      (OFF_CUR + 4*RP_STRIDE_I*4)       // 81920
#define KIDX_STRIDE_I 16384
#define OFF_PAIR      (OFF_KIDX + 4*KIDX_STRIDE_I*4)    // 344064
#define PAIR_STRIDE   16384                             // int2 per matrix

// ---------------- LDS layout of main kernel ----------------
#define SMEM_H_BYTES  (H_WIDTH*BTILE*4)                 // 229376
#define SMEM_STAGE    SMEM_H_BYTES
#define SMEM_BYTES    (SMEM_H_BYTES + NNZ_E*8)          // 294912 (< 320KB WGP LDS)

// ======================= preprocessing =======================

__global__ void k_zero(int* __restrict__ p, int n) {
    int i = blockIdx.x * blockDim.x + threadIdx.x;
    if (i < n) p[i] = 0;
}

__device__ __forceinline__ bool decode_nnz(int tid, int& m, int& k) {
    if (tid < NNZ_E)              { m = 0; k = tid;            return true; }
    if (tid < 2*NNZ_E)            { m = 1; k = tid - NNZ_E;    return true; }
    if (tid < 3*NNZ_E)            { m = 2; k = tid - 2*NNZ_E;  return true; }
    if (tid < 3*NNZ_E + NNZ_M)    { m = 3; k = tid - 3*NNZ_E;  return true; }
    return false;
}

__global__ void k_hist(const int* __restrict__ r0, const int* __restrict__ r1,
                       const int* __restrict__ r2, const int* __restrict__ rm,
                       int* __restrict__ ws_rp) {
    int tid = blockIdx.x * blockDim.x + threadIdx.x;
    int m, k;
    if (!decode_nnz(tid, m, k)) return;
    const int* rows = (m == 0) ? r0 : (m == 1) ? r1 : (m == 2) ? r2 : rm;
    atomicAdd(&ws_rp[m * RP_STRIDE_I + rows[k] + 1], 1);
}

__global__ void k_scan(int* __restrict__ ws_rp, int* __restrict__ ws_cur) {
    __shared__ int bufA[OUT_DIM + 1];
    __shared__ int bufB[OUT_DIM + 1];
    int m = blockIdx.x;
    int R = (m < 3) ? E_ROWS : OUT_DIM;
    int n = R + 1;
    int* rp  = ws_rp  + m * RP_STRIDE_I;
    int* cur = ws_cur + m * RP_STRIDE_I;
    for (int i = threadIdx.x; i < n; i += blockDim.x) bufA[i] = rp[i];
    __syncthreads();
    int* s = bufA; int* d = bufB;
    for (int off = 1; off < n; off <<= 1) {
        for (int i = threadIdx.x; i < n; i += blockDim.x)
            d[i] = s[i] + ((i >= off) ? s[i - off] : 0);
        __syncthreads();
        int* t = s; s = d; d = t;
    }
    for (int i = threadIdx.x; i < n; i += blockDim.x) { rp[i] = s[i]; cur[i] = s[i]; }
}

__global__ void k_scatter(const int* __restrict__ r0, const int* __restrict__ r1,
                          const int* __restrict__ r2, const int* __restrict__ rm,
                          int* __restrict__ ws_cur, int* __restrict__ ws_kidx) {
    int tid = blockIdx.x * blockDim.x + threadIdx.x;
    int m, k;
    if (!decode_nnz(tid, m, k)) return;
    const int* rows = (m == 0) ? r0 : (m == 1) ? r1 : (m == 2) ? r2 : rm;
    int pos = atomicAdd(&ws_cur[m * RP_STRIDE_I + rows[k]], 1);
    ws_kidx[m * KIDX_STRIDE_I + pos] = k;
}

// per-row selection sort by original nnz index -> deterministic summation order
__global__ void k_sortemit(const int* __restrict__ c0, const float* __restrict__ v0,
                           const int* __restrict__ c1, const float* __restrict__ v1,
                           const int* __restrict__ c2, const float* __restrict__ v2,
                           const int* __restrict__ cm, const float* __restrict__ vm,
                           const int* __restrict__ ws_rp, int* __restrict__ ws_kidx,
                           int2* __restrict__ ws_pair) {
    int tid = blockIdx.x * blockDim.x + threadIdx.x;
    int m, r;
    const int* cols; const float* vals;
    if      (tid < E_ROWS)            { m = 0; r = tid;              cols = c0; vals = v0; }
    else if (tid < 2*E_ROWS)          { m = 1; r = tid - E_ROWS;     cols = c1; vals = v1; }
    else if (tid < 3*E_ROWS)          { m = 2; r = tid - 2*E_ROWS;   cols = c2; vals = v2; }
    else if (tid < 3*E_ROWS+OUT_DIM)  { m = 3; r = tid - 3*E_ROWS;   cols = cm; vals = vm; }
    else return;
    const int* rp   = ws_rp   + m * RP_STRIDE_I;
    int*       kidx = ws_kidx + m * KIDX_STRIDE_I;
    int2*      pair = ws_pair + m * PAIR_STRIDE;
    int p0 = rp[r], p1 = rp[r + 1];
    for (int i = p0; i < p1 - 1; ++i) {
        int mn = i, kv = kidx[i];
        for (int j = i + 1; j < p1; ++j) { int kj = kidx[j]; if (kj < kv) { mn = j; kv = kj; } }
        if (mn != i) { kidx[mn] = kidx[i]; kidx[i] = kv; }
    }
    for (int p = p0; p < p1; ++p) {
        int k = kidx[p];
        pair[p] = make_int2(cols[k], __float_as_int(vals[k]));
    }
}

// ======================= main fused kernel =======================

typedef unsigned int u32x4 __attribute__((ext_vector_type(4)));
typedef int          i32x4 __attribute__((ext_vector_type(4)));
typedef int          i32x8 __attribute__((ext_vector_type(8)));

// TDM: 1-D tile of NNZ_E 8-byte {col,val} pairs, global -> LDS (per ISA cdna5 §8)
// clang-23 toolchain: 6-arg builtin (v4u g0, v8i g1, v4i g2, v4i g3, v8i extra, i32 cpol)
__device__ __forceinline__ void tdm_load_pairs(unsigned lds_byte_off, const int2* gsrc) {
    unsigned long long ga = (unsigned long long)(uintptr_t)gsrc;
    u32x4 g0;
    g0[0] = 1u;                                               // count=1, user mode
    g0[1] = lds_byte_off;                                     // lds_addr
    g0[2] = (unsigned)(ga & 0xFFFFFFFFull);                   // global_addr[31:0]
    g0[3] = (unsigned)((ga >> 32) & 0x01FFFFFFull) | 0x80000000u; // addr[56:32] | type=2
    i32x8 g1;
    g1[0] = 0x00030000;            // workgroup_mask=0, data_size=3 (8B), no pad/iterate
    g1[1] = (int)(NNZ_E << 16);    // tensor_dim0[15:0]=8192 in bits[31:16]
    g1[2] = 1 << 16;               // tensor_dim0[31:16]=0, tensor_dim1=1
    g1[3] = (int)(NNZ_E << 16);    // tile_dim0=8192 in bits[31:16]
    g1[4] = 1;                     // tile_dim1=1, tile_dim2=0
    g1[5] = NNZ_E;                 // tensor_dim0_stride[31:0]
    g1[6] = 0;                     // stride hi / dim1_stride lo
    g1[7] = 0;
    i32x4 gz4 = {0, 0, 0, 0};
    i32x8 gz8 = {0, 0, 0, 0, 0, 0, 0, 0};
    __builtin_amdgcn_tensor_load_to_lds(g0, g1, gz4, gz4, gz8, 0);
}

__device__ __forceinline__ int imax2(int a, int b) { return a > b ? a : b; }

// Process 4 consecutive sparse rows [r4, r4+4) for batch lane b.
// 4 independent acc chains -> 4x latency hiding at 2-waves/SIMD occupancy.
// Row bounds are uniform per 16-lane group, but the two 16-lane groups of a wave may diverge.
__device__ __forceinline__ void sparse_rows4(const int* __restrict__ rp, int r4,
                                             const int2* __restrict__ pairs,
                                             const float* __restrict__ h, int b,
                                             float acc[4]) {
    int e0 = rp[r4 + 0], e1 = rp[r4 + 1], e2 = rp[r4 + 2];
    int e3 = rp[r4 + 3], e4 = rp[r4 + 4];
    acc[0] = 0.f; acc[1] = 0.f; acc[2] = 0.f; acc[3] = 0.f;
    int mx = imax2(imax2(e1 - e0, e2 - e1), imax2(e3 - e2, e4 - e3));
    for (int i = 0; i < mx; ++i) {
        if (e0 + i < e1) { int2 cv = pairs[e0 + i]; acc[0] += __int_as_float(cv.y) * h[cv.x * BTILE + b]; }
        if (e1 + i < e2) { int2 cv = pairs[e1 + i]; acc[1] += __int_as_float(cv.y) * h[cv.x * BTILE + b]; }
        if (e2 + i < e3) { int2 cv = pairs[e2 + i]; acc[2] += __int_as_float(cv.y) * h[cv.x * BTILE + b]; }
        if (e3 + i < e4) { int2 cv = pairs[e3 + i]; acc[3] += __int_as_float(cv.y) * h[cv.x * BTILE + b]; }
    }
}

__global__ void __launch_bounds__(256, 1)
k_spmm(const float* __restrict__ x, float* __restrict__ out,
       const int* __restrict__ ws_rp, const int2* __restrict__ ws_pair) {
    extern __shared__ char smem[];
    float*      h     = (float*)smem;                         // h[f][16b], feature-major
    const int2* stage = (const int2*)(smem + SMEM_STAGE);     // 64KB pair staging
    const unsigned stage_lds = (unsigned)(uintptr_t)smem + SMEM_STAGE; // low 32b of generic = LDS offset

    const int t  = threadIdx.x;
    const int b  = t & 15;          // batch lane within tile (conflict-free banks)
    const int g  = t >> 4;          // row/feature group
    const int b0 = blockIdx.x * BTILE;

    // kick level-0 pair DMA early (one wave; TDM is a wave-level op)
    if (t < 32) tdm_load_pairs(stage_lds, ws_pair);

    // load x tile transposed: global float4 reads, conflict-free ds stores
    {
        const float* xr = x + (size_t)(b0 + b) * F0;
        for (int f0 = g * 4; f0 < F0; f0 += 64) {
            float4 v = *(const float4*)(xr + f0);
            h[(f0 + 0) * BTILE + b] = v.x;
            h[(f0 + 1) * BTILE + b] = v.y;
            h[(f0 + 2) * BTILE + b] = v.z;
            h[(f0 + 3) * BTILE + b] = v.w;
        }
    }
    __syncthreads();

    int w = F0;
    for (int lvl = 0; lvl < 3; ++lvl) {
        if (t < 32) __builtin_amdgcn_s_wait_tensorcnt(0);
        __syncthreads();                                   // staged pairs visible
        const int* rp = ws_rp + lvl * RP_STRIDE_I;
        for (int r4 = g * 4; r4 < E_ROWS; r4 += 64) {
            float acc[4];
            sparse_rows4(rp, r4, stage, h, b, acc);
            h[(w + r4 + 0) * BTILE + b] = acc[0];          // append e-level to h
            h[(w + r4 + 1) * BTILE + b] = acc[1];
            h[(w + r4 + 2) * BTILE + b] = acc[2];
            h[(w + r4 + 3) * BTILE + b] = acc[3];
        }
        w += E_ROWS;
        __syncthreads();                                   // staging fully consumed
        if (lvl < 2 && t < 32)
            tdm_load_pairs(stage_lds, ws_pair + (size_t)(lvl + 1) * PAIR_STRIDE);
    }

    // MAIN sparse layer: pairs streamed from L2 (uniform per group), float4 stores
    {
        const int*  rp = ws_rp   + 3 * RP_STRIDE_I;
        const int2* pm = ws_pair + 3 * PAIR_STRIDE;
        float* orow = out + (size_t)(b0 + b) * OUT_DIM;
        for (int r4 = g * 4; r4 < OUT_DIM; r4 += 64) {
            float acc[4];
            sparse_rows4(rp, r4, pm, h, b, acc);
            float4 st; st.x = acc[0]; st.y = acc[1]; st.z = acc[2]; st.w = acc[3];
            *(float4*)(orow + r4) = st;
        }
    }
}

// ======================= launch =======================

extern "C" void kernel_launch(void* const* d_in, const int* in_sizes, int n_in,
                              void* d_out, int out_size, void* d_ws, size_t ws_size,
                              hipStream_t stream) {
    (void)in_sizes; (void)n_in; (void)out_size; (void)ws_size;
    const float* x  = (const float*)d_in[0];
    const int*   r0 = (const int*)d_in[1];  const int* c0 = (const int*)d_in[2];  const float* v0 = (const float*)d_in[3];
    const int*   r1 = (const int*)d_in[4];  const int* c1 = (const int*)d_in[5];  const float* v1 = (const float*)d_in[6];
    const int*   r2 = (const int*)d_in[7];  const int* c2 = (const int*)d_in[8];  const float* v2 = (const float*)d_in[9];
    const int*   rm = (const int*)d_in[10]; const int* cm = (const int*)d_in[11]; const float* vm = (const float*)d_in[12];
    float* out = (float*)d_out;
    char*  ws  = (char*)d_ws;
    int*  ws_rp   = (int*)(ws + OFF_RP);
    int*  ws_cur  = (int*)(ws + OFF_CUR);
    int*  ws_kidx = (int*)(ws + OFF_KIDX);
    int2* ws_pair = (int2*)(ws + OFF_PAIR);

    (void)hipFuncSetAttribute((const void*)k_spmm,
                              hipFuncAttributeMaxDynamicSharedMemorySize, SMEM_BYTES);

    k_zero   <<<(OFF_KIDX/4 + 255)/256, 256, 0, stream>>>(ws_rp, OFF_KIDX/4);
    k_hist   <<<(3*NNZ_E + NNZ_M)/256, 256, 0, stream>>>(r0, r1, r2, rm, ws_rp);
    k_scan   <<<4, 1024, 0, stream>>>(ws_rp, ws_cur);
    k_scatter<<<(3*NNZ_E + NNZ_M)/256, 256, 0, stream>>>(r0, r1, r2, rm, ws_cur, ws_kidx);
    k_sortemit<<<(3*E_ROWS + OUT_DIM)/256, 256, 0, stream>>>(c0, v0, c1, v1, c2, v2, cm, vm,
                                                             ws_rp, ws_kidx, ws_pair);
    k_spmm   <<<B_ROWS/BTILE, 256, SMEM_BYTES, stream>>>(x, out, ws_rp, ws_pair);
}